// CODABlocks2D_9483287789757
// MI455X (gfx1250) — compile-verified
//
#include <hip/hip_runtime.h>

typedef float v8f __attribute__((ext_vector_type(8)));
typedef float v2f __attribute__((ext_vector_type(2)));

#define PI2F 6.28318530717958647692f

// ---------------- DFT basis tables (offsets in floats) ----------------
#define OFF_GC2   0L      // [128][96]  forward col basis, re|im packed, /16384
#define OFF_FCAT  12288L  // [128][128] forward row basis, rows 0..63 = cos, 64..127 = -sin
#define OFF_EWF   28672L  // [96][256]  inverse col basis full-res (Wo=128), Zr|Zi cols
#define OFF_EWH   53248L  // [96][128]  inverse col basis half-res (Wo=64)
#define OFF_FIRF  65536L  // [128][64]  inverse row cos full-res
#define OFF_FIIF  73728L  // [128][64]  inverse row -sin full-res
#define OFF_FIRH  81920L  // [64][64]   inverse row cos half-res
#define OFF_FIIH  86016L  // [64][64]   inverse row -sin half-res
#define TAB_N     90112

__global__ void init_tables(float* tab) {
  int i = blockIdx.x * blockDim.x + threadIdx.x;
  if (i >= TAB_N) return;
  int j = i;
  if (j < 12288) {                       // GC2: X[l] = (1/N) sum_w x[w] e^{-i 2pi l w/128}
    int w = j / 96, c = j % 96;
    int l = c % 48, im = c / 48;
    float v = 0.f;
    if (l < 33) {
      float th = PI2F * (float)((l * w) % 128) / 128.f;
      v = (im ? -sinf(th) : cosf(th)) * (1.f / 16384.f);
    }
    tab[OFF_GC2 + j] = v; return;
  }
  j -= 12288;
  if (j < 16384) {                       // FCAT: freq rows {0..31,96..127}
    int r = j / 128, h = j % 128;
    int rr = r & 63; int f = (rr < 32) ? rr : rr + 64;
    float th = PI2F * (float)((f * h) % 128) / 128.f;
    tab[OFF_FCAT + j] = (r < 64) ? cosf(th) : -sinf(th);
    return;
  }
  j -= 16384;
  if (j < 24576) {                       // EWF: Wo=128, l<32, c_l = (l?2:1)
    int r = j / 256, c = j % 256;
    int l = r % 48; int yi = r / 48;
    int w = c % 128; int zi = c / 128;
    float v = 0.f;
    if (l < 32) {
      float cl = (l == 0) ? 1.f : 2.f;
      float th = PI2F * (float)((l * w) % 128) / 128.f;
      float cv = cosf(th), sv = sinf(th);
      v = cl * (zi ? (yi ? cv : sv) : (yi ? -sv : cv));
    }
    tab[OFF_EWF + j] = v; return;
  }
  j -= 24576;
  if (j < 12288) {                       // EWH: Wo=64, l<33, Nyquist l=32 gets c=1
    int r = j / 128, c = j % 128;
    int l = r % 48; int yi = r / 48;
    int w = c % 64; int zi = c / 64;
    float v = 0.f;
    if (l < 33) {
      float cl = (l == 0 || l == 32) ? 1.f : 2.f;
      float th = PI2F * (float)((l * w) % 64) / 64.f;
      float cv = cosf(th), sv = sinf(th);
      v = cl * (zi ? (yi ? cv : sv) : (yi ? -sv : cv));
    }
    tab[OFF_EWH + j] = v; return;
  }
  j -= 12288;
  if (j < 8192) {                        // FIRF
    int h = j / 64, k = j % 64; int f = (k < 32) ? k : k + 64;
    tab[OFF_FIRF + j] = cosf(PI2F * (float)((f * h) % 128) / 128.f); return;
  }
  j -= 8192;
  if (j < 8192) {                        // FIIF
    int h = j / 64, k = j % 64; int f = (k < 32) ? k : k + 64;
    tab[OFF_FIIF + j] = -sinf(PI2F * (float)((f * h) % 128) / 128.f); return;
  }
  j -= 8192;
  if (j < 4096) {                        // FIRH
    int h = j / 64, k = j % 64;
    tab[OFF_FIRH + j] = cosf(PI2F * (float)((k * h) % 64) / 64.f); return;
  }
  j -= 4096;
  {                                      // FIIH
    int h = j / 64, k = j % 64;
    tab[OFF_FIIH + j] = -sinf(PI2F * (float)((k * h) % 64) / 64.f);
  }
}

// ---------------- generic strided batched WMMA GEMM ----------------
// C[z] = alpha * A[z] @ B[z] (+ C[z] if acc).  base(z) = (z/div)*so + (z%div)*si
struct GP {
  const float* A; const float* B; float* C;
  int M, N, K;
  int a_rs, a_cs, b_rs, b_cs, c_rs, c_cs;
  int a_div; long a_so, a_si;
  int b_div; long b_so, b_si;
  int c_div; long c_so, c_si;
  float alpha; int acc;
};

__global__ __launch_bounds__(128) void gemm_wmma(GP p) {
  __shared__ float As[64][17];
  __shared__ float Bs[16][66];
  int z = blockIdx.z;
  const float* Ab = p.A + (long)(z / p.a_div) * p.a_so + (long)(z % p.a_div) * p.a_si;
  const float* Bb = p.B + (long)(z / p.b_div) * p.b_so + (long)(z % p.b_div) * p.b_si;
  float*       Cb = p.C + (long)(z / p.c_div) * p.c_so + (long)(z % p.c_div) * p.c_si;
  int m0 = blockIdx.y * 64, n0 = blockIdx.x * 64;
  int tid = threadIdx.x;
  int wave = tid >> 5, lane = tid & 31;
  int wm = (wave >> 1) * 32, wn = (wave & 1) * 32;
  int hi = lane >> 4, lo = lane & 15;

  const v8f vzero = {0.f,0.f,0.f,0.f,0.f,0.f,0.f,0.f};
  v8f acc[2][2];
#pragma unroll
  for (int i = 0; i < 2; i++)
#pragma unroll
    for (int j = 0; j < 2; j++) acc[i][j] = vzero;

  for (int k0 = 0; k0 < p.K; k0 += 16) {
#pragma unroll
    for (int t = 0; t < 8; t++) {        // A tile 64x16
      int idx = tid + t * 128;
      int am = idx >> 4, ak = idx & 15;
      int gm = m0 + am, gk = k0 + ak;
      float v = 0.f;
      if (gm < p.M && gk < p.K) v = Ab[(long)gm * p.a_rs + (long)gk * p.a_cs];
      As[am][ak] = v;
    }
#pragma unroll
    for (int t = 0; t < 8; t++) {        // B tile 16x64
      int idx = tid + t * 128;
      int bk = idx >> 6, bn = idx & 63;
      int gk = k0 + bk, gn = n0 + bn;
      float v = 0.f;
      if (gk < p.K && gn < p.N) v = Bb[(long)gk * p.b_rs + (long)gn * p.b_cs];
      Bs[bk][bn] = v;
    }
    __syncthreads();
#pragma unroll
    for (int kk = 0; kk < 16; kk += 4) {
      v2f af[2], bf[2];
#pragma unroll
      for (int i = 0; i < 2; i++) {      // A frag: 16x4, lanes 0-15 K{0,1}, 16-31 K{2,3}
        int m = wm + i * 16 + lo;
        af[i].x = As[m][kk + 2 * hi];
        af[i].y = As[m][kk + 2 * hi + 1];
      }
#pragma unroll
      for (int jj = 0; jj < 2; jj++) {   // B frag: 4x16
        int n = wn + jj * 16 + lo;
        bf[jj].x = Bs[kk + 2 * hi][n];
        bf[jj].y = Bs[kk + 2 * hi + 1][n];
      }
#pragma unroll
      for (int i = 0; i < 2; i++)
#pragma unroll
        for (int jj = 0; jj < 2; jj++)
          acc[i][jj] = __builtin_amdgcn_wmma_f32_16x16x4_f32(
              false, af[i], false, bf[jj], (short)0, acc[i][jj], false, false);
    }
    __syncthreads();
  }
#pragma unroll
  for (int i = 0; i < 2; i++)
#pragma unroll
    for (int jj = 0; jj < 2; jj++) {
#pragma unroll
      for (int v = 0; v < 8; v++) {      // C: row = v + 8*hi, col = lo
        int r = m0 + wm + i * 16 + hi * 8 + v;
        int c = n0 + wn + jj * 16 + lo;
        if (r < p.M && c < p.N) {
          long off = (long)r * p.c_rs + (long)c * p.c_cs;
          float val = p.alpha * acc[i][jj][v];
          if (p.acc) val += Cb[off];
          Cb[off] = val;
        }
      }
    }
}

static void gemm(hipStream_t s, const float* A, const float* B, float* C,
                 int M, int N, int K,
                 int ars, int acs, int brs, int bcs, int crs, int ccs,
                 int nb,
                 int adiv, long aso, long asi,
                 int bdiv, long bso, long bsi,
                 int cdiv, long cso, long csi,
                 float alpha, int acc) {
  GP p{A, B, C, M, N, K, ars, acs, brs, bcs, crs, ccs,
       adiv, aso, asi, bdiv, bso, bsi, cdiv, cso, csi, alpha, acc};
  dim3 grid((N + 63) / 64, (M + 63) / 64, nb);
  gemm_wmma<<<grid, dim3(128), 0, s>>>(p);
}

// ---------------- elementwise / reduction kernels ----------------
// S[b][k][re|im] from T2[b][128][96]: Sr = top.re - bot.im, Si = top.im + bot.re
__global__ void combine_fwd(const float* T2, float* S, long total) {
  long i = (long)blockIdx.x * blockDim.x + threadIdx.x;
  if (i >= total) return;
  int c = (int)(i % 96); long r = i / 96;
  int k = (int)(r % 64); long b = r / 64;
  const float* Tb = T2 + b * 12288;
  int l = c % 48;
  float v;
  if (c < 48) v = Tb[k * 96 + l]      - Tb[(64 + k) * 96 + 48 + l];
  else        v = Tb[k * 96 + 48 + l] + Tb[(64 + k) * 96 + l];
  S[i] = v;
}

// K/Q/V spectrum assembly: modes 16 conv (+ spectral resample skip for K/Q), DC bias folded
__global__ void spec_kqv(const float* S2, const float* w, const float* ws,
                         const float* bs, float* OUT, int with_resample) {
  long i = (long)blockIdx.x * blockDim.x + threadIdx.x;
  if (i >= 2048L * 64 * 48) return;
  int l = (int)(i % 48); long r = i / 48;
  int k = (int)(r % 64); long r2 = r / 64;
  int ch = (int)(r2 % 32); int img = (int)(r2 / 32);
  const float* Sp = S2 + (long)img * 6144 + (long)k * 96;
  float xr = Sp[l], xi = Sp[48 + l];
  float sr = 0.f, si = 0.f;
  if (with_resample && l < 33) { float cc = ws[ch]; sr = cc * xr; si = cc * xi; }
  if (l < 16 && (k < 16 || k >= 48)) {
    int sgn = (k < 16) ? 0 : 1; int m = (k < 16) ? k : (k - 48);
    long wi = (long)sgn * 16384 + (long)ch * 512 + m * 32 + l * 2;
    float wr = w[wi], wim = w[wi + 1];
    sr += xr * wr - xi * wim;
    si += xr * wim + xi * wr;
  }
  if (k == 0 && l == 0) sr += bs[ch];
  float* Op = OUT + (((long)img * 32 + ch) * 64 + k) * 96;
  Op[l] = sr; Op[48 + l] = si;
}

// mixer spectrum: single channel, modes 32
__global__ void spec_m(const float* S2, const float* w, float* OUT) {
  long i = (long)blockIdx.x * blockDim.x + threadIdx.x;
  if (i >= 64L * 64 * 48) return;
  int l = (int)(i % 48); long r = i / 48;
  int k = (int)(r % 64); int img = (int)(r / 64);
  float sr = 0.f, si = 0.f;
  if (l < 32) {
    const float* Sp = S2 + (long)img * 6144 + (long)k * 96;
    float xr = Sp[l], xi = Sp[48 + l];
    int sgn = (k < 32) ? 0 : 1; int m = (k < 32) ? k : (k - 32);
    long wi = (long)sgn * 2048 + m * 64 + l * 2;
    sr = xr * w[wi] - xi * w[wi + 1];
    si = xr * w[wi + 1] + xi * w[wi];
  }
  float* Op = OUT + ((long)img * 64 + k) * 96;
  Op[l] = sr; Op[48 + l] = si;
}

// P projection: contract 32 head-channels per mode, DC bias folded
__global__ void contract_p(const float* S2P, const float* wP, const float* bPs, float* OUT) {
  long i = (long)blockIdx.x * blockDim.x + threadIdx.x;
  if (i >= 64L * 64 * 48) return;
  int l = (int)(i % 48); long r = i / 48;
  int k = (int)(r % 64); int bt = (int)(r / 64);
  float sr = 0.f, si = 0.f;
  if (l < 32) {
    int sgn = (k < 32) ? 0 : 1; int m = (k < 32) ? k : (k - 32);
    for (int c = 0; c < 32; c++) {
      const float* Sp = S2P + (((long)bt * 32 + c) * 64 + k) * 96;
      float xr = Sp[l], xi = Sp[48 + l];
      long wi = (long)sgn * 65536 + (long)c * 2048 + m * 64 + l * 2;
      sr += xr * wP[wi] - xi * wP[wi + 1];
      si += xr * wP[wi + 1] + xi * wP[wi];
    }
  }
  if (k == 0 && l == 0) sr += bPs[0];
  float* Op = OUT + ((long)bt * 64 + k) * 96;
  Op[l] = sr; Op[48 + l] = si;
}

__global__ void skip_v_k(float* VO, const float* XN, const float* wVs) {
  long i = (long)blockIdx.x * blockDim.x + threadIdx.x;
  if (i >= 2048L * 16384) return;
  int p = (int)(i & 16383);
  long img2 = i >> 14;
  int ch = (int)(img2 & 31); long img = img2 >> 5;
  VO[i] += wVs[ch] * XN[img * 16384 + p];
}

__global__ void skip_p_k(float* PRJ, const float* AOT, const float* wPs) {
  long i = (long)blockIdx.x * blockDim.x + threadIdx.x;
  if (i >= 64L * 16384) return;
  long bt = i >> 14; int p = (int)(i & 16383);
  float s = 0.f;
  for (int c = 0; c < 32; c++) s += wPs[c] * AOT[((bt * 32 + c) << 14) + p];
  PRJ[i] += s;
}

// InstanceNorm with optional pre-add, affine-skip, bias and exact-erf GELU
__global__ __launch_bounds__(256) void inorm_k(
    const float* X, const float* ADD, const float* SKIP,
    const float* skw, const float* skb,
    const float* Gv, const float* Bv, int gi,
    float* OUT, int HW, int gelu_flag) {
  __shared__ float rs[256], rq[256];
  long base = (long)blockIdx.x * HW;
  float s = 0.f, s2 = 0.f;
  for (int p = threadIdx.x; p < HW; p += 256) {
    float v = X[base + p];
    if (ADD) v += ADD[base + p];
    s += v; s2 += v * v;
  }
  rs[threadIdx.x] = s; rq[threadIdx.x] = s2;
  __syncthreads();
  for (int o = 128; o > 0; o >>= 1) {
    if (threadIdx.x < (unsigned)o) { rs[threadIdx.x] += rs[threadIdx.x + o]; rq[threadIdx.x] += rq[threadIdx.x + o]; }
    __syncthreads();
  }
  float mean = rs[0] / (float)HW;
  float var  = rq[0] / (float)HW - mean * mean;
  float inv  = rsqrtf(var + 1e-5f);
  float g = Gv[gi], b = Bv[gi];
  float sw = skw ? skw[0] : 1.f;
  float sb = skb ? skb[0] : 0.f;
  for (int p = threadIdx.x; p < HW; p += 256) {
    float v = X[base + p];
    if (ADD) v += ADD[base + p];
    float y = (v - mean) * inv * g + b;
    if (SKIP) y += sw * SKIP[base + p] + sb;
    if (gelu_flag) y = 0.5f * y * (1.f + erff(y * 0.70710678118654752f));
    OUT[base + p] = y;
  }
}

__global__ void softmax32(float* S) {
  __shared__ float sh[32];
  float* p = S + (long)blockIdx.x * 32;
  float v = p[threadIdx.x];
  sh[threadIdx.x] = v; __syncthreads();
  for (int o = 16; o > 0; o >>= 1) {
    if (threadIdx.x < (unsigned)o) sh[threadIdx.x] = fmaxf(sh[threadIdx.x], sh[threadIdx.x + o]);
    __syncthreads();
  }
  float mx = sh[0]; __syncthreads();
  float e = __expf(v - mx);
  sh[threadIdx.x] = e; __syncthreads();
  for (int o = 16; o > 0; o >>= 1) {
    if (threadIdx.x < (unsigned)o) sh[threadIdx.x] += sh[threadIdx.x + o];
    __syncthreads();
  }
  p[threadIdx.x] = e / sh[0];
}

static inline int blocks(long n) { return (int)((n + 255) / 256); }

// ---------------- orchestration ----------------
extern "C" void kernel_launch(void* const* d_in, const int* in_sizes, int n_in,
                              void* d_out, int out_size, void* d_ws, size_t ws_size,
                              hipStream_t stream) {
  (void)in_sizes; (void)n_in; (void)out_size; (void)ws_size;
  const float* x    = (const float*)d_in[0];
  const float* wK   = (const float*)d_in[1];
  const float* wKs  = (const float*)d_in[2];
  const float* bKs  = (const float*)d_in[3];
  const float* wQ   = (const float*)d_in[4];
  const float* wQs  = (const float*)d_in[5];
  const float* bQs  = (const float*)d_in[6];
  const float* wV   = (const float*)d_in[7];
  const float* wVs  = (const float*)d_in[8];
  const float* bVs  = (const float*)d_in[9];
  const float* wP   = (const float*)d_in[10];
  const float* wPs  = (const float*)d_in[11];
  const float* bPs  = (const float*)d_in[12];
  const float* wM0  = (const float*)d_in[13];
  const float* wM0s = (const float*)d_in[14];
  const float* bM0s = (const float*)d_in[15];
  const float* wM1  = (const float*)d_in[16];
  const float* wM1s = (const float*)d_in[17];
  const float* bM1s = (const float*)d_in[18];
  const float* ng   = (const float*)d_in[19];
  const float* nbv  = (const float*)d_in[20];
  float* out = (float*)d_out;

  float* W    = (float*)d_ws;
  float* TAB  = W;
  float* XN   = W + 131072;
  float* S2   = XN + 1048576;
  float* T1   = S2 + 393216;
  float* T2   = T1 + 786432;
  float* SPEC = T2 + 786432;
  float* Z    = SPEC + 12582912;
  float* KO   = Z + 33554432;
  float* QO   = KO + 8388608;
  float* VO   = QO + 8388608;
  float* SC   = VO + 33554432;
  float* AOT  = SC + 65536;
  float* S2P  = AOT + 33554432;
  // aliases (phase-disjoint reuse)
  float* T1P = VO;
  float* T2P = Z;
  float* PRJ = KO;
  float* ATT = KO + 1048576;
  float* AN  = KO + 2097152;
  float* MF  = KO + 3145728;
  float* MO  = KO + 4194304;
  float* M1O = KO + 5242880;

  init_tables<<<blocks(TAB_N), 256, 0, stream>>>(TAB);

  // norm1: xn = inorm(x; g0,b0)
  inorm_k<<<64, 256, 0, stream>>>(x, nullptr, nullptr, nullptr, nullptr, ng, nbv, 0, XN, 16384, 0);

  // forward truncated spectrum of xn: T1 = xn@Gc, T2 = Fcat@T1, combine -> S2
  gemm(stream, XN, TAB + OFF_GC2, T1, 128, 96, 128, 128, 1, 96, 1, 96, 1, 64,
       1, 16384, 0, 1, 0, 0, 1, 12288, 0, 1.f, 0);
  gemm(stream, TAB + OFF_FCAT, T1, T2, 128, 96, 128, 128, 1, 96, 1, 96, 1, 64,
       1, 0, 0, 1, 12288, 0, 1, 12288, 0, 1.f, 0);
  combine_fwd<<<blocks(64L * 64 * 96), 256, 0, stream>>>(T2, S2, 64L * 64 * 96);

  // ---- K (half-res, spectral-resample skip folded) ----
  spec_kqv<<<blocks(2048L * 64 * 48), 256, 0, stream>>>(S2, wK, wKs, bKs, SPEC, 1);
  gemm(stream, SPEC, TAB + OFF_EWH, Z, 64, 128, 96, 96, 1, 128, 1, 128, 1, 2048,
       1, 6144, 0, 1, 0, 0, 1, 8192, 0, 1.f, 0);
  gemm(stream, TAB + OFF_FIRH, Z, KO, 64, 64, 64, 64, 1, 128, 1, 64, 1, 2048,
       1, 0, 0, 1, 8192, 0, 1, 4096, 0, 1.f, 0);
  gemm(stream, TAB + OFF_FIIH, Z + 64, KO, 64, 64, 64, 64, 1, 128, 1, 64, 1, 2048,
       1, 0, 0, 1, 8192, 0, 1, 4096, 0, 1.f, 1);

  // ---- Q ----
  spec_kqv<<<blocks(2048L * 64 * 48), 256, 0, stream>>>(S2, wQ, wQs, bQs, SPEC, 1);
  gemm(stream, SPEC, TAB + OFF_EWH, Z, 64, 128, 96, 96, 1, 128, 1, 128, 1, 2048,
       1, 6144, 0, 1, 0, 0, 1, 8192, 0, 1.f, 0);
  gemm(stream, TAB + OFF_FIRH, Z, QO, 64, 64, 64, 64, 1, 128, 1, 64, 1, 2048,
       1, 0, 0, 1, 8192, 0, 1, 4096, 0, 1.f, 0);
  gemm(stream, TAB + OFF_FIIH, Z + 64, QO, 64, 64, 64, 64, 1, 128, 1, 64, 1, 2048,
       1, 0, 0, 1, 8192, 0, 1, 4096, 0, 1.f, 1);

  // ---- V (full-res, spatial skip) ----
  spec_kqv<<<blocks(2048L * 64 * 48), 256, 0, stream>>>(S2, wV, nullptr, bVs, SPEC, 0);
  gemm(stream, SPEC, TAB + OFF_EWF, Z, 64, 256, 96, 96, 1, 256, 1, 256, 1, 2048,
       1, 6144, 0, 1, 0, 0, 1, 16384, 0, 1.f, 0);
  gemm(stream, TAB + OFF_FIRF, Z, VO, 128, 128, 64, 64, 1, 256, 1, 128, 1, 2048,
       1, 0, 0, 1, 16384, 0, 1, 16384, 0, 1.f, 0);
  gemm(stream, TAB + OFF_FIIF, Z + 128, VO, 128, 128, 64, 64, 1, 256, 1, 128, 1, 2048,
       1, 0, 0, 1, 16384, 0, 1, 16384, 0, 1.f, 1);
  skip_v_k<<<blocks(2048L * 16384), 256, 0, stream>>>(VO, XN, wVs);

  // ---- attention: scores = Q K^T / 64, softmax, out = A V (direct into [bt][head][hw]) ----
  gemm(stream, QO, KO, SC, 32, 32, 4096, 131072, 1, 1, 131072, 32, 1, 64,
       32, 4194304, 4096, 32, 4194304, 4096, 1, 1024, 0, 1.f / 64.f, 0);
  softmax32<<<2048, 32, 0, stream>>>(SC);
  gemm(stream, SC, VO, AOT, 32, 16384, 32, 32, 1, 524288, 1, 524288, 1, 64,
       1, 1024, 0, 32, 16777216, 16384, 32, 16777216, 16384, 1.f, 0);

  // ---- P projection: forward spectra of all 2048 head-images, contract, inverse, spatial skip ----
  gemm(stream, AOT, TAB + OFF_GC2, T1P, 128, 96, 128, 128, 1, 96, 1, 96, 1, 2048,
       1, 16384, 0, 1, 0, 0, 1, 12288, 0, 1.f, 0);
  gemm(stream, TAB + OFF_FCAT, T1P, T2P, 128, 96, 128, 128, 1, 96, 1, 96, 1, 2048,
       1, 0, 0, 1, 12288, 0, 1, 12288, 0, 1.f, 0);
  combine_fwd<<<blocks(2048L * 64 * 96), 256, 0, stream>>>(T2P, S2P, 2048L * 64 * 96);
  contract_p<<<blocks(64L * 64 * 48), 256, 0, stream>>>(S2P, wP, bPs, SPEC);
  gemm(stream, SPEC, TAB + OFF_EWF, Z, 64, 256, 96, 96, 1, 256, 1, 256, 1, 64,
       1, 6144, 0, 1, 0, 0, 1, 16384, 0, 1.f, 0);
  gemm(stream, TAB + OFF_FIRF, Z, PRJ, 128, 128, 64, 64, 1, 256, 1, 128, 1, 64,
       1, 0, 0, 1, 16384, 0, 1, 16384, 0, 1.f, 0);
  gemm(stream, TAB + OFF_FIIF, Z + 128, PRJ, 128, 128, 64, 64, 1, 256, 1, 128, 1, 64,
       1, 0, 0, 1, 16384, 0, 1, 16384, 0, 1.f, 1);
  skip_p_k<<<blocks(64L * 16384), 256, 0, stream>>>(PRJ, AOT, wPs);

  // attention = inorm(projd + x; g1,b1); an = inorm(attention; g2,b2)
  inorm_k<<<64, 256, 0, stream>>>(PRJ, x, nullptr, nullptr, nullptr, ng, nbv, 1, ATT, 16384, 0);
  inorm_k<<<64, 256, 0, stream>>>(ATT, nullptr, nullptr, nullptr, nullptr, ng, nbv, 2, AN, 16384, 0);

  // ---- mixer layer 0: gelu( inorm(specconv(an); g3,b3) + wM0s*an + bM0s ) ----
  gemm(stream, AN, TAB + OFF_GC2, T1, 128, 96, 128, 128, 1, 96, 1, 96, 1, 64,
       1, 16384, 0, 1, 0, 0, 1, 12288, 0, 1.f, 0);
  gemm(stream, TAB + OFF_FCAT, T1, T2, 128, 96, 128, 128, 1, 96, 1, 96, 1, 64,
       1, 0, 0, 1, 12288, 0, 1, 12288, 0, 1.f, 0);
  combine_fwd<<<blocks(64L * 64 * 96), 256, 0, stream>>>(T2, S2, 64L * 64 * 96);
  spec_m<<<blocks(64L * 64 * 48), 256, 0, stream>>>(S2, wM0, SPEC);
  gemm(stream, SPEC, TAB + OFF_EWF, Z, 64, 256, 96, 96, 1, 256, 1, 256, 1, 64,
       1, 6144, 0, 1, 0, 0, 1, 16384, 0, 1.f, 0);
  gemm(stream, TAB + OFF_FIRF, Z, MF, 128, 128, 64, 64, 1, 256, 1, 128, 1, 64,
       1, 0, 0, 1, 16384, 0, 1, 16384, 0, 1.f, 0);
  gemm(stream, TAB + OFF_FIIF, Z + 128, MF, 128, 128, 64, 64, 1, 256, 1, 128, 1, 64,
       1, 0, 0, 1, 16384, 0, 1, 16384, 0, 1.f, 1);
  inorm_k<<<64, 256, 0, stream>>>(MF, nullptr, AN, wM0s, bM0s, ng, nbv, 3, MO, 16384, 1);

  // ---- mixer layer 1 ----
  gemm(stream, MO, TAB + OFF_GC2, T1, 128, 96, 128, 128, 1, 96, 1, 96, 1, 64,
       1, 16384, 0, 1, 0, 0, 1, 12288, 0, 1.f, 0);
  gemm(stream, TAB + OFF_FCAT, T1, T2, 128, 96, 128, 128, 1, 96, 1, 96, 1, 64,
       1, 0, 0, 1, 12288, 0, 1, 12288, 0, 1.f, 0);
  combine_fwd<<<blocks(64L * 64 * 96), 256, 0, stream>>>(T2, S2, 64L * 64 * 96);
  spec_m<<<blocks(64L * 64 * 48), 256, 0, stream>>>(S2, wM1, SPEC);
  gemm(stream, SPEC, TAB + OFF_EWF, Z, 64, 256, 96, 96, 1, 256, 1, 256, 1, 64,
       1, 6144, 0, 1, 0, 0, 1, 16384, 0, 1.f, 0);
  gemm(stream, TAB + OFF_FIRF, Z, MF, 128, 128, 64, 64, 1, 256, 1, 128, 1, 64,
       1, 0, 0, 1, 16384, 0, 1, 16384, 0, 1.f, 0);
  gemm(stream, TAB + OFF_FIIF, Z + 128, MF, 128, 128, 64, 64, 1, 256, 1, 128, 1, 64,
       1, 0, 0, 1, 16384, 0, 1, 16384, 0, 1.f, 1);
  inorm_k<<<64, 256, 0, stream>>>(MF, nullptr, MO, wM1s, bM1s, ng, nbv, 4, M1O, 16384, 0);

  // output = inorm(m; g5,b5) + attention
  inorm_k<<<64, 256, 0, stream>>>(M1O, nullptr, ATT, nullptr, nullptr, ng, nbv, 5, out, 16384, 0);
}